// GlobalGraph_40724879901231
// MI455X (gfx1250) — compile-verified
//
#include <hip/hip_runtime.h>
#include <hip/hip_bf16.h>
#include <math.h>

typedef __attribute__((ext_vector_type(2))) float v2f;
typedef __attribute__((ext_vector_type(8))) float v8f;

#define D 128
#define LDA 132   // padded LDS stride (floats)

// ---------------------------------------------------------------------------
// GEMM: H = X @ W   (X: [N,128], W: [128,128], H: [N,128])
// block = 256 threads = 8 waves; block tile = 16 rows x 128 cols;
// each wave owns a 16x16 tile, K-loop of 32 x V_WMMA_F32_16X16X4_F32.
// ---------------------------------------------------------------------------
__global__ __launch_bounds__(256) void gemm_h_kernel(
    const float* __restrict__ X, const float* __restrict__ W,
    float* __restrict__ H, int N) {
  __shared__ float As[16 * LDA];
  const int row0 = blockIdx.x * 16;
  const int tid  = threadIdx.x;

  // stage A tile (16x128) into LDS, coalesced float4 loads
  for (int i = tid; i < 16 * (D / 4); i += 256) {
    int r = (i * 4) / D, c = (i * 4) % D;
    int gr = row0 + r;
    float4 v = make_float4(0.f, 0.f, 0.f, 0.f);
    if (gr < N) v = *(const float4*)(X + (long long)gr * D + c);
    *(float4*)(As + r * LDA + c) = v;
  }
  __syncthreads();

  const int wave = tid / 32, lane = tid % 32;
  const int lh = lane / 16, lm = lane % 16;
  const int col = wave * 16 + lm;

  v8f acc = {};
  for (int k = 0; k < D; k += 4) {
    const int ka = k + lh * 2;
    v2f a = *(const v2f*)(As + lm * LDA + ka);   // A[lm][ka..ka+1]
    v2f b;
    b.x = W[(ka + 0) * D + col];                 // B[ka][col]
    b.y = W[(ka + 1) * D + col];                 // B[ka+1][col]
    if (k + 8 < D) __builtin_prefetch(&W[(k + 8) * D + col], 0, 0);
    acc = __builtin_amdgcn_wmma_f32_16x16x4_f32(
        /*neg_a=*/false, a, /*neg_b=*/false, b,
        /*c_mod=*/(short)0, acc, /*reuse_a=*/false, /*reuse_b=*/false);
  }
  // D layout: VGPR v -> M = v + 8*lh, N = lm
  for (int v = 0; v < 8; ++v) {
    int gr = row0 + v + 8 * lh;
    if (gr < N) H[(long long)gr * D + col] = acc[v];
  }
}

// ---------------------------------------------------------------------------
// a_src[row] = h[row]·att_src ; a_dst[row] = h[row]·att_dst   (1 wave / row)
// ---------------------------------------------------------------------------
__global__ __launch_bounds__(256) void rowdot_kernel(
    const float* __restrict__ h, const float* __restrict__ att_s,
    const float* __restrict__ att_d, float* __restrict__ asrc,
    float* __restrict__ adst, int N) {
  int row = blockIdx.x * 8 + threadIdx.x / 32;
  if (row >= N) return;
  int lane = threadIdx.x % 32;
  float4 hv = *(const float4*)(h + (long long)row * D + lane * 4);
  float4 s4 = *(const float4*)(att_s + lane * 4);
  float4 d4 = *(const float4*)(att_d + lane * 4);
  float ss = hv.x * s4.x + hv.y * s4.y + hv.z * s4.z + hv.w * s4.w;
  float dd = hv.x * d4.x + hv.y * d4.y + hv.z * d4.z + hv.w * d4.w;
  for (int off = 16; off > 0; off >>= 1) {
    ss += __shfl_xor(ss, off, 32);
    dd += __shfl_xor(dd, off, 32);
  }
  if (lane == 0) { asrc[row] = ss; adst[row] = dd; }
}

// ---------------------------------------------------------------------------
// init: h_gat = b_gat (broadcast), agg = 0, m = -inf, z = 0, cnt = 0
// ---------------------------------------------------------------------------
__global__ __launch_bounds__(256) void init_kernel(
    float* __restrict__ hgat, float* __restrict__ agg, float* __restrict__ m,
    float* __restrict__ z, float* __restrict__ cnt,
    const float* __restrict__ bgat, int N) {
  long long i = (long long)blockIdx.x * 256 + threadIdx.x;
  if (i >= (long long)N * D) return;
  int f = (int)(i & (D - 1));
  hgat[i] = bgat[f];
  agg[i]  = 0.f;
  if (f == 0) {
    int n = (int)(i >> 7);
    m[n] = -__builtin_inff();
    z[n] = 0.f;
    cnt[n] = 0.f;
  }
}

// ---------------------------------------------------------------------------
// phase 1: e = leaky_relu(a_src[src] + a_dst[dst]); segment max via atomics
// edges e<E come from edge_index, e>=E are self loops.
// ---------------------------------------------------------------------------
__global__ __launch_bounds__(256) void edge_max_kernel(
    const int* __restrict__ src, const int* __restrict__ dst,
    const float* __restrict__ asrc, const float* __restrict__ adst,
    float* __restrict__ ebuf, float* __restrict__ m, int E, int N) {
  int e = blockIdx.x * 256 + threadIdx.x;
  if (e >= E + N) return;
  int s = (e < E) ? src[e] : (e - E);
  int t = (e < E) ? dst[e] : (e - E);
  float v = asrc[s] + adst[t];
  v = (v > 0.f) ? v : 0.2f * v;      // leaky_relu, slope 0.2
  ebuf[e] = v;
  atomicMax(m + t, v);               // global_atomic_max_num_f32
}

// ---------------------------------------------------------------------------
// phase 2: p = exp(e - m[dst]); z[dst] += p  (p overwrites ebuf)
// ---------------------------------------------------------------------------
__global__ __launch_bounds__(256) void edge_expsum_kernel(
    const int* __restrict__ dst, float* __restrict__ ebuf,
    const float* __restrict__ m, float* __restrict__ z, int E, int N) {
  int e = blockIdx.x * 256 + threadIdx.x;
  if (e >= E + N) return;
  int t = (e < E) ? dst[e] : (e - E);
  float p = __expf(ebuf[e] - m[t]);
  ebuf[e] = p;
  atomicAdd(z + t, p);
}

// ---------------------------------------------------------------------------
// phase 3: h_gat[dst] += h[src] * (p / z[dst])   (one thread per (edge,feat))
// ---------------------------------------------------------------------------
__global__ __launch_bounds__(256) void edge_scatter_kernel(
    const int* __restrict__ src, const int* __restrict__ dst,
    const float* __restrict__ ebuf, const float* __restrict__ z,
    const float* __restrict__ h, float* __restrict__ hgat, int E, int N) {
  long long i = (long long)blockIdx.x * 256 + threadIdx.x;
  if (i >= (long long)(E + N) * D) return;
  int f = (int)(i & (D - 1));
  int e = (int)(i >> 7);
  int s = (e < E) ? src[e] : (e - E);
  int t = (e < E) ? dst[e] : (e - E);
  float alpha = ebuf[e] / z[t];
  atomicAdd(hgat + (long long)t * D + f, h[(long long)s * D + f] * alpha);
}

// ---------------------------------------------------------------------------
// SAGE mean-agg numerators: agg[dst0] += h_gat[src0]; cnt[dst0] += 1
// (original edges only, no self loops)
// ---------------------------------------------------------------------------
__global__ __launch_bounds__(256) void sage_agg_kernel(
    const int* __restrict__ src, const int* __restrict__ dst,
    const float* __restrict__ hgat, float* __restrict__ agg,
    float* __restrict__ cnt, int E) {
  long long i = (long long)blockIdx.x * 256 + threadIdx.x;
  if (i >= (long long)E * D) return;
  int f = (int)(i & (D - 1));
  int e = (int)(i >> 7);
  int s = src[e], t = dst[e];
  atomicAdd(agg + (long long)t * D + f, hgat[(long long)s * D + f]);
  if (f == 0) atomicAdd(cnt + t, 1.0f);
}

// ---------------------------------------------------------------------------
// out = L2norm( (agg/max(cnt,1)) @ W_l + b_l + h_gat @ W_r )
// same WMMA tiling; block owns all 128 cols of its 16 rows, so the row
// normalization is done in LDS in the same kernel.
// ---------------------------------------------------------------------------
__global__ __launch_bounds__(256) void gemm_out_kernel(
    const float* __restrict__ AGG, const float* __restrict__ CNT,
    const float* __restrict__ HG, const float* __restrict__ Wl,
    const float* __restrict__ bl, const float* __restrict__ Wr,
    float* __restrict__ OUT, int N) {
  __shared__ float A1[16 * LDA];
  __shared__ float A2[16 * LDA];
  __shared__ float O [16 * LDA];
  const int row0 = blockIdx.x * 16;
  const int tid  = threadIdx.x;

  for (int i = tid; i < 16 * (D / 4); i += 256) {
    int r = (i * 4) / D, c = (i * 4) % D;
    int gr = row0 + r;
    float4 v1 = make_float4(0.f, 0.f, 0.f, 0.f);
    float4 v2 = v1;
    if (gr < N) {
      float invc = 1.0f / fmaxf(CNT[gr], 1.0f);
      float4 a = *(const float4*)(AGG + (long long)gr * D + c);
      v1 = make_float4(a.x * invc, a.y * invc, a.z * invc, a.w * invc);
      v2 = *(const float4*)(HG + (long long)gr * D + c);
    }
    *(float4*)(A1 + r * LDA + c) = v1;
    *(float4*)(A2 + r * LDA + c) = v2;
  }
  __syncthreads();

  const int wave = tid / 32, lane = tid % 32;
  const int lh = lane / 16, lm = lane % 16;
  const int col = wave * 16 + lm;

  v8f acc = {};
  for (int k = 0; k < D; k += 4) {
    const int ka = k + lh * 2;
    v2f a = *(const v2f*)(A1 + lm * LDA + ka);
    v2f b;
    b.x = Wl[(ka + 0) * D + col];
    b.y = Wl[(ka + 1) * D + col];
    acc = __builtin_amdgcn_wmma_f32_16x16x4_f32(false, a, false, b,
                                                (short)0, acc, false, false);
  }
  for (int k = 0; k < D; k += 4) {
    const int ka = k + lh * 2;
    v2f a = *(const v2f*)(A2 + lm * LDA + ka);
    v2f b;
    b.x = Wr[(ka + 0) * D + col];
    b.y = Wr[(ka + 1) * D + col];
    acc = __builtin_amdgcn_wmma_f32_16x16x4_f32(false, a, false, b,
                                                (short)0, acc, false, false);
  }
  const float bias = bl[col];
  for (int v = 0; v < 8; ++v) O[(v + 8 * lh) * LDA + col] = acc[v] + bias;
  __syncthreads();

  // L2 normalize each of the 16 rows; wave w handles rows 2w, 2w+1
  for (int rr = 0; rr < 2; ++rr) {
    int r = wave * 2 + rr;
    int gr = row0 + r;
    float s = 0.f;
    for (int c = lane; c < D; c += 32) { float x = O[r * LDA + c]; s += x * x; }
    for (int off = 16; off > 0; off >>= 1) s += __shfl_xor(s, off, 32);
    float inv = 1.0f / fmaxf(sqrtf(s), 1e-12f);
    if (gr < N)
      for (int c = lane; c < D; c += 32)
        OUT[(long long)gr * D + c] = O[r * LDA + c] * inv;
  }
}

// ---------------------------------------------------------------------------
extern "C" void kernel_launch(void* const* d_in, const int* in_sizes, int n_in,
                              void* d_out, int out_size, void* d_ws, size_t ws_size,
                              hipStream_t stream) {
  const float* x    = (const float*)d_in[0];
  const int*   edge = (const int*)  d_in[1];
  const float* Wg   = (const float*)d_in[2];
  const float* atts = (const float*)d_in[3];
  const float* attd = (const float*)d_in[4];
  const float* bg   = (const float*)d_in[5];
  const float* Wl   = (const float*)d_in[6];
  const float* bl   = (const float*)d_in[7];
  const float* Wr   = (const float*)d_in[8];
  float* out = (float*)d_out;

  const int N = in_sizes[0] / D;      // 100000
  const int E = in_sizes[1] / 2;      // 600000
  const int* src = edge;
  const int* dst = edge + E;

  // workspace partition (floats)
  float* h    = (float*)d_ws;         // N*128
  float* hgat = h    + (long long)N * D;
  float* agg  = hgat + (long long)N * D;
  float* asrc = agg  + (long long)N * D;
  float* adst = asrc + N;
  float* m    = adst + N;
  float* z    = m    + N;
  float* cnt  = z    + N;
  float* ebuf = cnt  + N;             // E+N

  const int T = 256;
  long long nd      = (long long)N * D;
  long long en_d    = (long long)(E + N) * D;
  long long e_d     = (long long)E * D;

  gemm_h_kernel<<<(N + 15) / 16, T, 0, stream>>>(x, Wg, h, N);
  rowdot_kernel<<<(N + 7) / 8, T, 0, stream>>>(h, atts, attd, asrc, adst, N);
  init_kernel<<<(unsigned)((nd + T - 1) / T), T, 0, stream>>>(hgat, agg, m, z, cnt, bg, N);
  edge_max_kernel<<<(E + N + T - 1) / T, T, 0, stream>>>(src, dst, asrc, adst, ebuf, m, E, N);
  edge_expsum_kernel<<<(E + N + T - 1) / T, T, 0, stream>>>(dst, ebuf, m, z, E, N);
  edge_scatter_kernel<<<(unsigned)((en_d + T - 1) / T), T, 0, stream>>>(src, dst, ebuf, z, h, hgat, E, N);
  sage_agg_kernel<<<(unsigned)((e_d + T - 1) / T), T, 0, stream>>>(src, dst, hgat, agg, cnt, E);
  gemm_out_kernel<<<(N + 15) / 16, T, 0, stream>>>(agg, cnt, hgat, Wl, bl, Wr, out, N);
}